// MemoryBankV_88476326297767
// MI455X (gfx1250) — compile-verified
//
#include <hip/hip_runtime.h>

typedef __attribute__((ext_vector_type(2))) float v2f;
typedef __attribute__((ext_vector_type(8))) float v8f;

#define C_CLASSES 12
#define FEAT 32
#define ALPHA 0.9f

#define WAVES_PER_BLOCK 4
#define THREADS_PER_BLOCK (WAVES_PER_BLOCK * 32)
#define MAX_NBLK_PER_BATCH 128
// partial layout per block: [2 halves][8 vgprs][32 lanes] raw D + [32 lanes] raw cnt
#define PARTIAL_FLOATS (2 * 8 * 32 + 32)   // 544

// ---------------------------------------------------------------------------
// Kernel 1: streaming segment-sum via V_WMMA_F32_16X16X4_F32.
//   D[feat, cls] += A[feat, k] * B[k, cls]
//   A: 16 features x 4 voxels (f32), contiguous voxel pairs per lane
//   B: 4 voxels x 16 classes one-hot, built in registers from labels (exact)
// 2-step manual unroll: 8 voxels / iteration, 4 WMMAs, 8 global loads issued
// together for memory-level parallelism.
// ---------------------------------------------------------------------------
__global__ __launch_bounds__(THREADS_PER_BLOCK)
void segsum_wmma_kernel(const float* __restrict__ emb,
                        const int*   __restrict__ lab,
                        float*       __restrict__ partial,
                        long long N,              // voxels per batch
                        int nblk_per_batch)
{
    // force wave id scalar so the main loop bounds live in SGPRs
    const int wave = __builtin_amdgcn_readfirstlane((int)(threadIdx.x >> 5));
    const int lane = threadIdx.x & 31;
    const int batch      = blockIdx.x / nblk_per_batch;
    const int blkInBatch = blockIdx.x % nblk_per_batch;

    const float* E = emb + (long long)batch * FEAT * N;
    const int*   L = lab + (long long)batch * N;

    const int waves_per_batch = nblk_per_batch * WAVES_PER_BLOCK;
    const int w = blkInBatch * WAVES_PER_BLOCK + wave;

    const long long steps_total    = N >> 2;  // 4 voxels per step
    const long long steps_per_wave = (steps_total + waves_per_batch - 1) / waves_per_batch;
    long long s0 = (long long)w * steps_per_wave;
    long long s1 = s0 + steps_per_wave;
    if (s1 > steps_total) s1 = steps_total;
    if (s0 > steps_total) s0 = steps_total;

    const int fA0   = lane & 15;     // feature row for this lane (first half)
    const int hi    = lane >> 4;     // 0: K={0,1}, 1: K={2,3}
    const int kbase = hi * 2;
    const int myc   = lane & 15;     // class column this lane provides in B

    // running per-lane stream pointers
    const float* pA0 = E + (long long)fA0 * N        + s0 * 4 + kbase; // feats 0..15
    const float* pA1 = E + (long long)(fA0 + 16) * N + s0 * 4 + kbase; // feats 16..31
    const int*   pL  = L + s0 * 4 + kbase;

    v8f d0 = {};   // seg_sum for features 0..15  x classes 0..15
    v8f d1 = {};   // seg_sum for features 16..31 x classes 0..15
    float cnt = 0.0f;

    long long nsteps = s1 - s0;
    long long npair  = nsteps >> 1;

    for (long long i = 0; i < npair; ++i) {
        // WGP-scope prefetch ~512B ahead on both feature-row streams
        __builtin_prefetch(pA0 + 128, 0, 3);
        __builtin_prefetch(pA1 + 128, 0, 3);

        // issue all global loads for 8 voxels up front
        v2f a0 = *(const v2f*)(pA0);
        v2f a1 = *(const v2f*)(pA1);
        v2f a2 = *(const v2f*)(pA0 + 4);
        v2f a3 = *(const v2f*)(pA1 + 4);
        int l0 = pL[0];
        int l1 = pL[1];
        int l2 = pL[4];
        int l3 = pL[5];

        int c0 = (l0 > 10) ? (C_CLASSES - 1) : l0;
        int c1 = (l1 > 10) ? (C_CLASSES - 1) : l1;
        int c2 = (l2 > 10) ? (C_CLASSES - 1) : l2;
        int c3 = (l3 > 10) ? (C_CLASSES - 1) : l3;

        v2f b01, b23;
        b01.x = (c0 == myc) ? 1.0f : 0.0f;
        b01.y = (c1 == myc) ? 1.0f : 0.0f;
        b23.x = (c2 == myc) ? 1.0f : 0.0f;
        b23.y = (c3 == myc) ? 1.0f : 0.0f;
        cnt += b01.x + b01.y + b23.x + b23.y;

        d0 = __builtin_amdgcn_wmma_f32_16x16x4_f32(
                 false, a0, false, b01, (short)0, d0, false, false);
        d1 = __builtin_amdgcn_wmma_f32_16x16x4_f32(
                 false, a1, false, b01, (short)0, d1, false, false);
        d0 = __builtin_amdgcn_wmma_f32_16x16x4_f32(
                 false, a2, false, b23, (short)0, d0, false, false);
        d1 = __builtin_amdgcn_wmma_f32_16x16x4_f32(
                 false, a3, false, b23, (short)0, d1, false, false);

        pA0 += 8; pA1 += 8; pL += 8;
    }

    if (nsteps & 1) {   // tail: one 4-voxel step
        v2f a0 = *(const v2f*)(pA0);
        v2f a1 = *(const v2f*)(pA1);
        int l0 = pL[0];
        int l1 = pL[1];
        int c0 = (l0 > 10) ? (C_CLASSES - 1) : l0;
        int c1 = (l1 > 10) ? (C_CLASSES - 1) : l1;
        v2f bb;
        bb.x = (c0 == myc) ? 1.0f : 0.0f;
        bb.y = (c1 == myc) ? 1.0f : 0.0f;
        cnt += bb.x + bb.y;
        d0 = __builtin_amdgcn_wmma_f32_16x16x4_f32(
                 false, a0, false, bb, (short)0, d0, false, false);
        d1 = __builtin_amdgcn_wmma_f32_16x16x4_f32(
                 false, a1, false, bb, (short)0, d1, false, false);
    }

    // ---- deterministic block reduction via LDS (fixed order over waves) ----
    __shared__ float redD[WAVES_PER_BLOCK][2][8][32];
    __shared__ float redC[WAVES_PER_BLOCK][32];

    #pragma unroll
    for (int j = 0; j < 8; ++j) {
        redD[wave][0][j][lane] = d0[j];
        redD[wave][1][j][lane] = d1[j];
    }
    redC[wave][lane] = cnt;
    __syncthreads();

    float* pout = partial + (long long)blockIdx.x * PARTIAL_FLOATS;
    for (int idx = threadIdx.x; idx < 512; idx += THREADS_PER_BLOCK) {
        float ssum = 0.0f;
        #pragma unroll
        for (int wv = 0; wv < WAVES_PER_BLOCK; ++wv)
            ssum += (&redD[wv][0][0][0])[idx];
        pout[idx] = ssum;
    }
    if (threadIdx.x < 32) {
        float csum = 0.0f;
        #pragma unroll
        for (int wv = 0; wv < WAVES_PER_BLOCK; ++wv)
            csum += redC[wv][threadIdx.x];
        pout[512 + threadIdx.x] = csum;
    }
}

// ---------------------------------------------------------------------------
// Kernel 2: reduce per-block partials (fixed order), per-class mean, then the
// sequential per-batch EMA, exactly like the reference scan.
// One thread per (class, feature) output element.
// ---------------------------------------------------------------------------
__global__ __launch_bounds__(C_CLASSES * FEAT)
void finalize_kernel(const float* __restrict__ partial,
                     const float* __restrict__ proto_in,
                     float*       __restrict__ proto_out,
                     int nblk_per_batch,
                     int nbatch)
{
    const int t = threadIdx.x;
    if (t >= C_CLASSES * FEAT) return;
    const int c = t / FEAT;
    const int f = t % FEAT;

    // invert the raw D layout: feat = 16*half + j + 8*hi, class = lane&15
    const int half = f >> 4;
    const int hi   = (f >> 3) & 1;
    const int j    = f & 7;
    const int lane = c + 16 * hi;
    const int rawD = half * 256 + j * 32 + lane;

    float p = proto_in[t];
    for (int b = 0; b < nbatch; ++b) {
        const float* base = partial + (long long)b * nblk_per_batch * PARTIAL_FLOATS;
        float ssum = 0.0f, cn = 0.0f;
        for (int blk = 0; blk < nblk_per_batch; ++blk) {
            const float* pp = base + (long long)blk * PARTIAL_FLOATS;
            ssum += pp[rawD];
            cn   += pp[512 + c] + pp[512 + c + 16];
        }
        if (cn > 0.0f) {
            const float mean = ssum / fmaxf(cn, 1.0f);
            p = ALPHA * p + (1.0f - ALPHA) * mean;
        }
    }
    proto_out[t] = p;
}

extern "C" void kernel_launch(void* const* d_in, const int* in_sizes, int n_in,
                              void* d_out, int out_size, void* d_ws, size_t ws_size,
                              hipStream_t stream)
{
    const float* emb   = (const float*)d_in[0];  // [B, 32, N] f32
    const int*   lab   = (const int*)d_in[1];    // [B, N] i32
    const float* proto = (const float*)d_in[2];  // [12, 32] f32
    float*       out   = (float*)d_out;          // [12, 32] f32
    float*       ws    = (float*)d_ws;

    const int nbatch = 2;
    const long long N = (long long)in_sizes[1] / nbatch;   // voxels per batch

    // size grid to available scratch (one 544-float slot per block)
    int nblk = MAX_NBLK_PER_BATCH;
    const size_t slot_bytes = (size_t)PARTIAL_FLOATS * sizeof(float);
    size_t max_blocks = ws_size / slot_bytes;
    if ((size_t)(nbatch * nblk) > max_blocks) {
        nblk = (int)(max_blocks / nbatch);
        if (nblk < 1) nblk = 1;
    }

    segsum_wmma_kernel<<<dim3(nbatch * nblk), dim3(THREADS_PER_BLOCK), 0, stream>>>(
        emb, lab, ws, N, nblk);

    finalize_kernel<<<dim3(1), dim3(C_CLASSES * FEAT), 0, stream>>>(
        ws, proto, out, nblk, nbatch);
}